// ConstantB0T2_46273977647339
// MI455X (gfx1250) — compile-verified
//
#include <hip/hip_runtime.h>
#include <math.h>

typedef __attribute__((ext_vector_type(2))) float v2f;
typedef __attribute__((ext_vector_type(8))) float v8f;

#define M_TOTAL    82080
#define NSAMP_SHOT 2565      // NS * OSF: samples per shot
#define N_IMG      128
#define NCH        4
#define MTILE      32        // k-space samples per block (two 16-wide N-tiles)
#define LDS_STRIDE 132       // 128 + 4 pad -> conflict-free B-frag loads

__global__ __launch_bounds__(256, 2)
void ConstantB0T2_nufft_wmma(const float* __restrict__ img,
                             const float* __restrict__ traj,
                             float* __restrict__ out)
{
    // Phase tables, stored [n][y] padded: Eyc = cos(gy*ky), Eyn = -sin(gy*ky)
    __shared__ float Eyc[MTILE * LDS_STRIDE];
    __shared__ float Eyn[MTILE * LDS_STRIDE];
    __shared__ float Exc[MTILE * LDS_STRIDE];
    __shared__ float Exn[MTILE * LDS_STRIDE];
    __shared__ float red[NCH][MTILE][2];   // per-(channel, n) complex reduction

    const int tid = threadIdx.x;
    const int m0  = blockIdx.x * MTILE;    // this block's 32 k-space samples

    // ---- Step A: build Ey / Ex tables (21M sincos across grid: trivial) ----
    for (int i = tid; i < N_IMG * MTILE; i += 256) {
        int y = i >> 5;          // spatial index (same grid for x and y dims)
        int n = i & (MTILE - 1); // sample within tile
        float kyv = traj[M_TOTAL + m0 + n];   // trajectory row 1 = ky
        float kxv = traj[m0 + n];             // trajectory row 0 = kx
        float g = (float)(y - 64);            // gx/gy = arange(128) - 64
        float sy, cy, sx, cx;
        sincosf(g * kyv, &sy, &cy);
        sincosf(g * kxv, &sx, &cx);
        Eyc[n * LDS_STRIDE + y] = cy;
        Eyn[n * LDS_STRIDE + y] = -sy;        // Ey = cos - i sin
        Exc[n * LDS_STRIDE + y] = cx;
        Exn[n * LDS_STRIDE + y] = -sx;        // Ex = cos - i sin
    }
    if (tid < NCH * MTILE * 2) ((float*)red)[tid] = 0.0f;
    __syncthreads();

    const int lane  = tid & 31;
    const int wave  = tid >> 5;     // 8 wave32 per block
    const int lhalf = lane >> 4;    // which 16-lane half
    const int l15   = lane & 15;

    // B fragments: column n lives in lane l15 (n-tile 0: n=l15, n-tile 1: n=16+l15)
    const float* bre0 = &Eyc[l15 * LDS_STRIDE];
    const float* bim0 = &Eyn[l15 * LDS_STRIDE];
    const float* bre1 = &Eyc[(16 + l15) * LDS_STRIDE];
    const float* bim1 = &Eyn[(16 + l15) * LDS_STRIDE];

    // A rows: this wave owns x = 16*wave .. 16*wave+15 (lane = row, both halves)
    const int xrow = 16 * wave + l15;

    for (int j = 0; j < NCH; ++j) {   // channel loop (Ey shared across channels)
        const float* arow = img + j * (N_IMG * N_IMG) + xrow * N_IMG;
        v8f acc_re0 = {0.f,0.f,0.f,0.f,0.f,0.f,0.f,0.f};
        v8f acc_im0 = {0.f,0.f,0.f,0.f,0.f,0.f,0.f,0.f};
        v8f acc_re1 = {0.f,0.f,0.f,0.f,0.f,0.f,0.f,0.f};
        v8f acc_im1 = {0.f,0.f,0.f,0.f,0.f,0.f,0.f,0.f};

        // ---- Step B: 16x128x32 f32 GEMM via V_WMMA_F32_16X16X4_F32 ----
        // One A-fragment load feeds 4 WMMAs -> halved image/L2 traffic.
#pragma unroll
        for (int k0 = 0; k0 < N_IMG; k0 += 4) {
            int kk = k0 + 2 * lhalf;                 // K half per A/B layout
            v2f a   = *(const v2f*)(arow + kk);      // per-lane image row pair
            v2f br0 = *(const v2f*)(bre0 + kk);      // conflict-free LDS
            v2f bi0 = *(const v2f*)(bim0 + kk);
            v2f br1 = *(const v2f*)(bre1 + kk);
            v2f bi1 = *(const v2f*)(bim1 + kk);
            acc_re0 = __builtin_amdgcn_wmma_f32_16x16x4_f32(
                false, a, false, br0, (short)0, acc_re0, false, false);
            acc_im0 = __builtin_amdgcn_wmma_f32_16x16x4_f32(
                false, a, false, bi0, (short)0, acc_im0, false, false);
            acc_re1 = __builtin_amdgcn_wmma_f32_16x16x4_f32(
                false, a, false, br1, (short)0, acc_re1, false, false);
            acc_im1 = __builtin_amdgcn_wmma_f32_16x16x4_f32(
                false, a, false, bi1, (short)0, acc_im1, false, false);
        }

        // ---- Step C: fused stage 2: multiply by Ex and reduce over x ----
        float pre0 = 0.f, pim0 = 0.f, pre1 = 0.f, pim1 = 0.f;
#pragma unroll
        for (int i = 0; i < 8; ++i) {
            // C/D layout: VGPR i -> matrix row i + 8*lhalf, column l15 (per N-tile)
            int x = 16 * wave + 8 * lhalf + i;
            float exc0 = Exc[l15 * LDS_STRIDE + x];
            float exn0 = Exn[l15 * LDS_STRIDE + x];        // = -sin
            float exc1 = Exc[(16 + l15) * LDS_STRIDE + x];
            float exn1 = Exn[(16 + l15) * LDS_STRIDE + x];
            float ar0 = acc_re0[i], ai0 = acc_im0[i];
            float ar1 = acc_re1[i], ai1 = acc_im1[i];
            pre0 += ar0 * exc0 - ai0 * exn0;   // (ar+i*ai)*(exc+i*exn)
            pim0 += ar0 * exn0 + ai0 * exc0;
            pre1 += ar1 * exc1 - ai1 * exn1;
            pim1 += ar1 * exn1 + ai1 * exc1;
        }
        atomicAdd(&red[j][l15][0], pre0);        // ds_add_f32
        atomicAdd(&red[j][l15][1], pim0);
        atomicAdd(&red[j][16 + l15][0], pre1);
        atomicAdd(&red[j][16 + l15][1], pim1);
    }
    __syncthreads();

    // ---- Step D: B0/T2 multiplier + store (complex64 interleaved) ----
    if (tid < NCH * MTILE) {
        int c = tid >> 5;
        int n = tid & (MTILE - 1);
        int m = m0 + n;
        float t = 2.0e-6f * (float)(m % NSAMP_SHOT);  // dwell 0.002ms -> seconds
        float decay = __expf(-t * 12.5f);             // 1/T2 = 1/0.08
        float s, cc;
        __sincosf(157.07963267948966f * t, &s, &cc);  // 2*pi*B0, B0=25Hz
        float mr = decay * cc;
        float mi = decay * s;
        float re = red[c][n][0];
        float im = red[c][n][1];
        out[(size_t)(c * M_TOTAL + m) * 2 + 0] = re * mr - im * mi;
        out[(size_t)(c * M_TOTAL + m) * 2 + 1] = re * mi + im * mr;
    }
}

extern "C" void kernel_launch(void* const* d_in, const int* in_sizes, int n_in,
                              void* d_out, int out_size, void* d_ws, size_t ws_size,
                              hipStream_t stream) {
    (void)in_sizes; (void)n_in; (void)out_size; (void)d_ws; (void)ws_size;
    const float* img  = (const float*)d_in[0];   // [1,4,128,128] f32
    const float* traj = (const float*)d_in[1];   // [2, 82080] f32
    float* out = (float*)d_out;                  // [1,4,82080] complex64 -> f32 pairs
    dim3 grid(M_TOTAL / MTILE);                  // 2565 blocks, no tail
    ConstantB0T2_nufft_wmma<<<grid, 256, 0, stream>>>(img, traj, out);
}